// HungarianLoss_24068996727203
// MI455X (gfx1250) — compile-verified
//
#include <hip/hip_runtime.h>
#include <math.h>

#define B_ 256
#define N_ 64
#define D_ 1024

typedef __attribute__((ext_vector_type(2))) float v2f;
typedef __attribute__((ext_vector_type(8))) float v8f;

// ---------------------------------------------------------------------------
// Kernel 1: batched Euclidean distance matrix via fp32 WMMA.
// One workgroup (512 thr = 16 waves) per batch; wave w computes 16x16 tile
// (w/4, w%4) of the 64x64 matrix P·T^T with V_WMMA_F32_16X16X4_F32, K=1024.
// ---------------------------------------------------------------------------
__global__ __launch_bounds__(512) void cdist_wmma_kernel(
    const float* __restrict__ pred, const float* __restrict__ target,
    float* __restrict__ cost) {
  const int b   = blockIdx.x;
  const int tid = threadIdx.x;
  const float* P = pred   + (size_t)b * N_ * D_;
  const float* T = target + (size_t)b * N_ * D_;

  __shared__ float p2s[N_], t2s[N_];
  __shared__ float red[512];

  // ---- per-row squared norms (fixed-order, deterministic) ----
  {
    const int row = tid >> 3, part = tid & 7;
    const float* pr = P + row * D_;
    float s = 0.f;
    for (int k = part; k < D_; k += 8) { float x = pr[k]; s += x * x; }
    red[tid] = s;
    __syncthreads();
    if (part == 0) {
      float tot = 0.f;
      for (int j = 0; j < 8; ++j) tot += red[(row << 3) + j];
      p2s[row] = tot;
    }
    __syncthreads();
    const float* tr = T + row * D_;
    s = 0.f;
    for (int k = part; k < D_; k += 8) { float x = tr[k]; s += x * x; }
    red[tid] = s;
    __syncthreads();
    if (part == 0) {
      float tot = 0.f;
      for (int j = 0; j < 8; ++j) tot += red[(row << 3) + j];
      t2s[row] = tot;
    }
    __syncthreads();
  }

  // ---- WMMA main loop ----
  const int wave = tid >> 5;
  const int lane = tid & 31;
  const int half = lane >> 4;   // 0: K=k0..k0+1, 1: K=k0+2..k0+3
  const int l16  = lane & 15;
  const int tm   = wave >> 2, tn = wave & 3;

  const float* pa = P + (tm * 16 + l16) * D_ + 2 * half;  // A: rows of pred
  const float* pb = T + (tn * 16 + l16) * D_ + 2 * half;  // B: rows of target (= cols of T^T)

  v8f acc = {0.f, 0.f, 0.f, 0.f, 0.f, 0.f, 0.f, 0.f};
#pragma unroll 8
  for (int k0 = 0; k0 < D_; k0 += 4) {
    v2f a  = *(const v2f*)(pa + k0);
    v2f bb = *(const v2f*)(pb + k0);
    acc = __builtin_amdgcn_wmma_f32_16x16x4_f32(
        /*neg_a=*/false, a, /*neg_b=*/false, bb,
        /*c_mod=*/(short)0, acc, /*reuse_a=*/false, /*reuse_b=*/false);
  }

  // C/D layout: VGPR r -> M = r (lanes 0-15) or M = 8+r (lanes 16-31), N = lane%16
  float* crow = cost + (size_t)b * N_ * N_;
  const int col = tn * 16 + l16;
#pragma unroll
  for (int r = 0; r < 8; ++r) {
    const int row = tm * 16 + r + 8 * half;
    float d2 = p2s[row] + t2s[col] - 2.f * acc[r];
    crow[row * N_ + col] = sqrtf(fmaxf(d2, 0.f));
  }
}

// ---------------------------------------------------------------------------
// Kernel 2: Hungarian (Kuhn-Munkres with potentials), one wave32 per batch.
// The 64x64 cost matrix is staged into LDS once (coalesced float4 loads) so
// the serial inner loop's dependent row reads are ds_load latency, not L2.
// Each lane owns columns ja=2*lane+1, jb=2*lane+2 (1-indexed, col 0 virtual).
// Argmin via shfl_xor butterfly, first-minimum tie-break (matches np.argmin).
// ---------------------------------------------------------------------------
__global__ __launch_bounds__(32) void hungarian_kernel(
    const float* __restrict__ cost, int* __restrict__ colfor) {
  const int b = blockIdx.x;
  const int lane = threadIdx.x;
  const float* C = cost + (size_t)b * N_ * N_;

  __shared__ float Cs[N_ * N_];                 // 16 KB cost cache
  __shared__ float u[N_ + 1], v[N_ + 1], minv[N_ + 1];
  __shared__ int   p[N_ + 1], way[N_ + 1], used[N_ + 1];

  // stage cost matrix into LDS: 1024 float4 across 32 lanes
  {
    const float4* src = (const float4*)C;
    float4* dst = (float4*)Cs;
    for (int idx = lane; idx < (N_ * N_) / 4; idx += 32) dst[idx] = src[idx];
  }
  for (int j = lane; j <= N_; j += 32) { u[j] = 0.f; v[j] = 0.f; p[j] = 0; way[j] = 0; }
  __syncthreads();

  const float INFV = 3.0e38f;
  const int ja = 2 * lane + 1;
  const int jb = 2 * lane + 2;

  for (int i = 1; i <= N_; ++i) {
    if (lane == 0) p[0] = i;
    for (int j = lane; j <= N_; j += 32) { minv[j] = INFV; used[j] = 0; }
    __syncthreads();

    int j0 = 0;
    while (true) {
      if (lane == 0) used[j0] = 1;
      __syncthreads();

      const int   i0  = p[j0];          // uniform broadcast read
      const float ui0 = u[i0];
      const float* crow = Cs + (i0 - 1) * N_;

      // relax reduced costs for the two owned columns
      if (!used[ja]) {
        float cur = crow[ja - 1] - ui0 - v[ja];
        if (cur < minv[ja]) { minv[ja] = cur; way[ja] = j0; }
      }
      if (!used[jb]) {
        float cur = crow[jb - 1] - ui0 - v[jb];
        if (cur < minv[jb]) { minv[jb] = cur; way[jb] = j0; }
      }

      // argmin over free columns (tie -> smallest index, like np.argmin)
      float ca = used[ja] ? INFV : minv[ja];
      float cb = used[jb] ? INFV : minv[jb];
      float bestv; int bestj;
      if (ca <= cb) { bestv = ca; bestj = ja; } else { bestv = cb; bestj = jb; }
      for (int off = 16; off; off >>= 1) {
        float ov = __shfl_xor(bestv, off, 32);
        int   oj = __shfl_xor(bestj, off, 32);
        if (ov < bestv || (ov == bestv && oj < bestj)) { bestv = ov; bestj = oj; }
      }
      const float delta = bestv;
      const int   j1    = bestj;

      // update potentials (distinct LDS addresses per lane -> no atomics)
      if (used[ja]) { u[p[ja]] += delta; v[ja] -= delta; } else { minv[ja] -= delta; }
      if (used[jb]) { u[p[jb]] += delta; v[jb] -= delta; } else { minv[jb] -= delta; }
      if (lane == 0) u[p[0]] += delta;   // virtual column 0 carries current row i
      __syncthreads();

      j0 = j1;
      if (p[j0] == 0) break;             // found a free column
    }

    // augment along alternating path (serial walk, lane 0)
    __syncthreads();
    if (lane == 0) {
      int jj = j0;
      while (jj != 0) { int jp = way[jj]; p[jj] = p[jp]; jj = jp; }
    }
    __syncthreads();
  }

  // col_for_row[p[j]-1] = j-1
  colfor[b * N_ + (p[ja] - 1)] = ja - 1;
  colfor[b * N_ + (p[jb] - 1)] = jb - 1;
}

// ---------------------------------------------------------------------------
// Kernel 3: gathered double-MSE per batch, fixed-order float4 accumulation +
// fixed LDS tree reduction (deterministic, no float atomics).
// ---------------------------------------------------------------------------
__global__ __launch_bounds__(256) void loss_kernel(
    const float* __restrict__ pred,  const float* __restrict__ target,
    const float* __restrict__ spred, const float* __restrict__ starget,
    const int* __restrict__ colfor,  float* __restrict__ partial) {
  const int b = blockIdx.x, tid = threadIdx.x;
  const size_t base = (size_t)b * N_ * D_;
  float accv = 0.f;
  for (int r = 0; r < N_; ++r) {
    const int c = colfor[b * N_ + r];
    const float4* p4 = (const float4*)(pred    + base + (size_t)r * D_);
    const float4* t4 = (const float4*)(target  + base + (size_t)c * D_);
    const float4* s4 = (const float4*)(spred   + base + (size_t)r * D_);
    const float4* q4 = (const float4*)(starget + base + (size_t)c * D_);
    float4 a = p4[tid], bb = t4[tid], cc = s4[tid], dd = q4[tid];
    float e;
    e = a.x - bb.x; accv += e * e;  e = a.y - bb.y; accv += e * e;
    e = a.z - bb.z; accv += e * e;  e = a.w - bb.w; accv += e * e;
    e = cc.x - dd.x; accv += e * e; e = cc.y - dd.y; accv += e * e;
    e = cc.z - dd.z; accv += e * e; e = cc.w - dd.w; accv += e * e;
  }
  __shared__ float sm[256];
  sm[tid] = accv;
  __syncthreads();
  for (int st = 128; st; st >>= 1) {
    if (tid < st) sm[tid] += sm[tid + st];
    __syncthreads();
  }
  if (tid == 0) partial[b] = sm[0] * (1.0f / (float)(N_ * D_));
}

// ---------------------------------------------------------------------------
// Kernel 4: final deterministic reduction of 256 per-batch partials.
// ---------------------------------------------------------------------------
__global__ __launch_bounds__(256) void final_reduce_kernel(
    const float* __restrict__ partial, float* __restrict__ out) {
  const int tid = threadIdx.x;
  __shared__ float sm[256];
  sm[tid] = partial[tid];
  __syncthreads();
  for (int st = 128; st; st >>= 1) {
    if (tid < st) sm[tid] += sm[tid + st];
    __syncthreads();
  }
  if (tid == 0) out[0] = sm[0];
}

extern "C" void kernel_launch(void* const* d_in, const int* in_sizes, int n_in,
                              void* d_out, int out_size, void* d_ws, size_t ws_size,
                              hipStream_t stream) {
  const float* pred    = (const float*)d_in[0];
  const float* target  = (const float*)d_in[1];
  const float* spred   = (const float*)d_in[2];
  const float* starget = (const float*)d_in[3];
  float* out = (float*)d_out;

  // workspace layout: [cost: B*64*64 f32][colfor: B*64 i32][partial: B f32]
  char* ws = (char*)d_ws;
  float* cost    = (float*)ws;
  int*   colfor  = (int*)(ws + (size_t)B_ * N_ * N_ * sizeof(float));
  float* partial = (float*)(ws + (size_t)B_ * N_ * N_ * sizeof(float)
                               + (size_t)B_ * N_ * sizeof(int));

  cdist_wmma_kernel<<<B_, 512, 0, stream>>>(pred, target, cost);
  hungarian_kernel<<<B_, 32, 0, stream>>>(cost, colfor);
  loss_kernel<<<B_, 256, 0, stream>>>(pred, target, spred, starget, colfor, partial);
  final_reduce_kernel<<<1, 256, 0, stream>>>(partial, out);
}